// myModel_82746839924789
// MI455X (gfx1250) — compile-verified
//
#include <hip/hip_runtime.h>
#include <hip/hip_bf16.h>

// ---------------------------------------------------------------------------
// RouteNet-style GNN critic for MI455X (gfx1250), fp32 WMMA (16x16x4) path.
// ---------------------------------------------------------------------------

typedef __attribute__((ext_vector_type(2))) float v2f;
typedef __attribute__((ext_vector_type(8))) float v8f;

constexpr int N_  = 50000;   // links
constexpr int NP_ = 50016;   // padded row count (multiple of 32)
constexpr int E_  = 800000;  // edges
constexpr int D_  = 128;     // link state dim
constexpr int U_  = 256;     // readout units
constexpr int T_  = 8;       // message passing steps

// D = A*B + C, A:16x4 f32, B:4x16 f32, C/D:16x16 f32
__device__ __forceinline__ v8f wmma_f32(v2f a, v2f b, v8f c) {
    return __builtin_amdgcn_wmma_f32_16x16x4_f32(
        /*neg_a=*/false, a, /*neg_b=*/false, b,
        /*c_mod=*/(short)0, c, /*reuse_a=*/false, /*reuse_b=*/false);
}

__device__ __forceinline__ v8f zero8() { v8f z = {}; return z; }

// A fragment from LDS, row-major [16][stride] (stride even => 8B aligned b64).
__device__ __forceinline__ v2f load_a(const float* A, int stride, int k, int ln, int hi) {
    return *(const v2f*)(A + ln * stride + k + 2 * hi);
}

// B fragment from global, row-major [K][ldb]; col n0+ln; rows k+2*hi, k+2*hi+1.
__device__ __forceinline__ v2f load_b(const float* __restrict__ B, int ldb, int k, int n0,
                                      int ln, int hi) {
    const float* p = B + (size_t)(k + 2 * hi) * ldb + n0 + ln;
    v2f r; r.x = p[0]; r.y = p[ldb];
    return r;
}

// Cheap activations: v_exp_f32 / v_rcp_f32 / v_tanh_f32 (gfx1250), branch-free.
__device__ __forceinline__ float fast_sigmoid(float x) {
    return __builtin_amdgcn_rcpf(1.0f + __expf(-x));
}
__device__ __forceinline__ float fast_tanh(float x) {
#if __has_builtin(__builtin_amdgcn_tanhf)
    return __builtin_amdgcn_tanhf(x);
#else
    return 1.0f - 2.0f * __builtin_amdgcn_rcpf(1.0f + __expf(2.0f * x));
#endif
}
__device__ __forceinline__ float fast_selu(float x) {
    float e = __expf(fminf(x, 0.0f));
    float neg = 1.6732632423543772f * (e - 1.0f);
    return 1.0507009873554805f * (x > 0.0f ? x : neg);
}

// --------- async global->LDS staging (gfx1250), guarded fallback ------------
#if __has_builtin(__builtin_amdgcn_global_load_async_to_lds_b128)
#define ASYNC_LDS 1
#endif

#ifdef ASYNC_LDS
typedef int v4i __attribute__((vector_size(16)));
typedef __attribute__((address_space(1))) v4i gv4i;   // global int4
typedef __attribute__((address_space(3))) v4i lv4i;   // LDS int4
#endif

__device__ __forceinline__ void stage16(float* lds, const float* g) {
#ifdef ASYNC_LDS
    __builtin_amdgcn_global_load_async_to_lds_b128((gv4i*)g, (lv4i*)lds, 0, 0);
#else
    *(float4*)lds = *(const float4*)g;
#endif
}
__device__ __forceinline__ void async_stage_fence() {
#ifdef ASYNC_LDS
#if __has_builtin(__builtin_amdgcn_s_wait_asynccnt)
    __builtin_amdgcn_s_wait_asynccnt(0);
#else
    asm volatile("s_wait_asynccnt 0x0" ::: "memory");
#endif
#endif
}

// Dual row-tile GEMM accumulate: one shared B fragment feeds two 16-row tiles.
template<int K>
__device__ __forceinline__ void gemm2(const float* A, int strideA,
                                      const float* __restrict__ B, int ldb, int n0,
                                      int ln, int hi, v8f& c0, v8f& c1)
{
#pragma unroll 2
    for (int k = 0; k < K; k += 4) {
        v2f b  = load_b(B, ldb, k, n0, ln, hi);
        v2f a0 = load_a(A,                strideA, k, ln, hi);
        v2f a1 = load_a(A + 16 * strideA, strideA, k, ln, hi);
        c0 = wmma_f32(a0, b, c0);
        c1 = wmma_f32(a1, b, c1);
    }
}

// ---------------------------------------------------------------------------
// Kernel 1: msg = selu(concat(h[f], h[s]) @ W_msg + b_msg); agg[s] += msg
// Persistent blocks; W_msg staged once per block into per-wave LDS slices.
// Register-prefetch pipeline hides the 16-row gather behind the WMMA loop.
// Dynamic LDS: A[16][260] + WL[8][4096] + sdst[16]  (~147 KB).
// ---------------------------------------------------------------------------
constexpr int MSG_SMEM_FLOATS = 16 * 260 + 8 * 4096 + 16;

__global__ __launch_bounds__(256)
void k_msg(const float* __restrict__ h, const int* __restrict__ fidx,
           const int* __restrict__ sidx, const float* __restrict__ W,
           const float* __restrict__ bias, float* __restrict__ agg, int nTiles)
{
    extern __shared__ float smem[];
    float* A    = smem;                 // [16][260] gathered edge features
    float* WL   = smem + 16 * 260;      // 8 wave slices, each [128][16][2]
    int*   sdst = (int*)(WL + 8 * 4096);

    const int t    = threadIdx.x;
    const int lane = t & 31;
    const int wave = t >> 5;
    const int ln   = lane & 15;
    const int hi   = lane >> 4;
    const int n0   = wave * 16;
    const int r    = t & 15;
    const int k0   = (t >> 4) * 16;

    float4 pre[4];
    int    preS = 0;
    auto prefetch = [&](int tile) {
        const int e0 = tile * 16;
        const int fi = fidx[e0 + r];
        const int si = sidx[e0 + r];
        preS = si;
        const float* src = (k0 < 128) ? (h + (size_t)fi * D_ + k0)
                                      : (h + (size_t)si * D_ + (k0 - 128));
#pragma unroll
        for (int j = 0; j < 4; ++j) pre[j] = ((const float4*)src)[j];
    };

    // issue first gather early; it overlaps the W staging below
    if (blockIdx.x < (unsigned)nTiles) prefetch(blockIdx.x);

    // stage W (row-major [256][128]) into per-wave slices (coalesced reads)
    for (int idx = t; idx < 256 * 128; idx += 256) {
        const int k = idx >> 7, ng = idx & 127;
        WL[(ng >> 4) * 4096 + (k >> 1) * 32 + (ng & 15) * 2 + (k & 1)] = W[idx];
    }
    const float  bn = bias[n0 + ln];
    // loop-invariant per-lane fragment base pointers (affine DS offsets)
    const float* ap = A + ln * 260 + 2 * hi;
    const float* bp = WL + wave * 4096 + hi * 32 + ln * 2;

    for (int tile = blockIdx.x; tile < nTiles; tile += gridDim.x) {
        __syncthreads();   // W ready on first pass; A/sdst free to overwrite
        {   // commit the prefetched tile to LDS
            float* dst = A + r * 260 + k0;
#pragma unroll
            for (int j = 0; j < 4; ++j) ((float4*)dst)[j] = pre[j];
            if (t < 16) sdst[t] = preS;
        }
        __syncthreads();

        const int nxt = tile + gridDim.x;
        if (nxt < nTiles) prefetch(nxt);   // overlaps the WMMA loop below

        v8f acc = zero8();
#pragma unroll 4
        for (int k = 0; k < 256; k += 4) {
            v2f a = *(const v2f*)(ap + k);        // offset 4k bytes
            v2f b = *(const v2f*)(bp + k * 16);   // offset 64k bytes (<16-bit imm)
            acc = wmma_f32(a, b, acc);
        }

        // fused SELU + segment-sum scatter (agg stays L2-resident)
#pragma unroll
        for (int v = 0; v < 8; ++v) {
            const int m = v + 8 * hi;
            float msg = fast_selu(acc[v] + bn);
            atomicAdd(&agg[(size_t)sdst[m] * D_ + n0 + ln], msg);
        }
    }
}

// ---------------------------------------------------------------------------
// Kernel 2: GRU update (keras reset_after=True), in-place on h. 32 rows/block,
// gate-by-gate phases keep accumulator pressure at ~8 v8f. Async LDS staging.
// ---------------------------------------------------------------------------
__global__ __launch_bounds__(256)
void k_update(float* __restrict__ h, const float* __restrict__ agg,
              const float* __restrict__ Wk, const float* __restrict__ Wr,
              const float* __restrict__ bias /*2x384*/)
{
    __shared__ float Ax[32 * 132];
    __shared__ float Ah[32 * 132];

    const int t    = threadIdx.x;
    const int lane = t & 31;
    const int wave = t >> 5;
    const int ln   = lane & 15;
    const int hi   = lane >> 4;
    const int row0 = blockIdx.x * 32;

    {   // stage 32 rows x 128 of agg and h (async global->LDS when available)
        const int rr = t & 31;
        const int c0 = (t >> 5) * 16;
        const float* sx = agg + (size_t)(row0 + rr) * D_ + c0;
        const float* sh = h   + (size_t)(row0 + rr) * D_ + c0;
#pragma unroll
        for (int j = 0; j < 16; j += 4) {
            stage16(&Ax[rr * 132 + c0 + j], sx + j);
            stage16(&Ah[rr * 132 + c0 + j], sh + j);
        }
        async_stage_fence();
    }
    __syncthreads();

    const int jz = wave * 16;
    const int n  = jz + ln;
    v8f p0, p1, q0, q1, z0, z1, r0, r1;

    // ---- z gate
    p0 = zero8(); p1 = zero8(); q0 = zero8(); q1 = zero8();
    gemm2<128>(Ax, 132, Wk, 384, jz, ln, hi, p0, p1);
    gemm2<128>(Ah, 132, Wr, 384, jz, ln, hi, q0, q1);
    {
        const float b0 = bias[n], b1 = bias[384 + n];
#pragma unroll
        for (int v = 0; v < 8; ++v) {
            z0[v] = fast_sigmoid(p0[v] + b0 + q0[v] + b1);
            z1[v] = fast_sigmoid(p1[v] + b0 + q1[v] + b1);
        }
    }
    // ---- r gate
    p0 = zero8(); p1 = zero8(); q0 = zero8(); q1 = zero8();
    gemm2<128>(Ax, 132, Wk, 384, 128 + jz, ln, hi, p0, p1);
    gemm2<128>(Ah, 132, Wr, 384, 128 + jz, ln, hi, q0, q1);
    {
        const float b0 = bias[128 + n], b1 = bias[384 + 128 + n];
#pragma unroll
        for (int v = 0; v < 8; ++v) {
            r0[v] = fast_sigmoid(p0[v] + b0 + q0[v] + b1);
            r1[v] = fast_sigmoid(p1[v] + b0 + q1[v] + b1);
        }
    }
    // ---- candidate + blend
    p0 = zero8(); p1 = zero8(); q0 = zero8(); q1 = zero8();
    gemm2<128>(Ax, 132, Wk, 384, 256 + jz, ln, hi, p0, p1);
    gemm2<128>(Ah, 132, Wr, 384, 256 + jz, ln, hi, q0, q1);
    {
        const float b0 = bias[256 + n], b1 = bias[384 + 256 + n];
#pragma unroll
        for (int v = 0; v < 8; ++v) {
            const int m = v + 8 * hi;
            float hh0 = fast_tanh(p0[v] + b0 + r0[v] * (q0[v] + b1));
            float hh1 = fast_tanh(p1[v] + b0 + r1[v] * (q1[v] + b1));
            float ho0 = Ah[m * 132 + n];
            float ho1 = Ah[(m + 16) * 132 + n];
            h[(size_t)(row0 + m) * D_ + n]      = z0[v] * ho0 + (1.0f - z0[v]) * hh0;
            h[(size_t)(row0 + 16 + m) * D_ + n] = z1[v] * ho1 + (1.0f - z1[v]) * hh1;
        }
    }
}

// ---------------------------------------------------------------------------
// Kernel 3: readout GRU step, in-place on h_ro. 32 rows/block; wave owns cols
// [w*32, w*32+32) as two 16-col sub-slices; gate-by-gate phases.
// ---------------------------------------------------------------------------
__global__ __launch_bounds__(256)
void k_readout(float* __restrict__ hro, const float* __restrict__ x,
               const float* __restrict__ Wk /*128x768*/, const float* __restrict__ Wr /*256x768*/,
               const float* __restrict__ bias /*2x768*/)
{
    __shared__ float Ax[32 * 132];
    __shared__ float Ah[32 * 260];

    const int t    = threadIdx.x;
    const int lane = t & 31;
    const int wave = t >> 5;
    const int ln   = lane & 15;
    const int hi   = lane >> 4;
    const int row0 = blockIdx.x * 32;

    {   // stage 32x128 of x and 32x256 of hro (async global->LDS when available)
        const int rr  = t & 31;
        const int c0x = (t >> 5) * 16;
        const int c0h = (t >> 5) * 32;
        const float* sx = x   + (size_t)(row0 + rr) * D_ + c0x;
        const float* sh = hro + (size_t)(row0 + rr) * U_ + c0h;
#pragma unroll
        for (int j = 0; j < 16; j += 4)
            stage16(&Ax[rr * 132 + c0x + j], sx + j);
#pragma unroll
        for (int j = 0; j < 32; j += 4)
            stage16(&Ah[rr * 260 + c0h + j], sh + j);
        async_stage_fence();
    }
    __syncthreads();

#pragma unroll 1
    for (int sub = 0; sub < 2; ++sub) {
        const int j0 = wave * 32 + sub * 16;   // in [0, 256)
        const int n  = j0 + ln;
        v8f p0, p1, q0, q1, z0, z1, r0, r1;

        // ---- z gate
        p0 = zero8(); p1 = zero8(); q0 = zero8(); q1 = zero8();
        gemm2<128>(Ax, 132, Wk, 768, j0, ln, hi, p0, p1);
        gemm2<256>(Ah, 260, Wr, 768, j0, ln, hi, q0, q1);
        {
            const float b0 = bias[n], b1 = bias[768 + n];
#pragma unroll
            for (int v = 0; v < 8; ++v) {
                z0[v] = fast_sigmoid(p0[v] + b0 + q0[v] + b1);
                z1[v] = fast_sigmoid(p1[v] + b0 + q1[v] + b1);
            }
        }
        // ---- r gate
        p0 = zero8(); p1 = zero8(); q0 = zero8(); q1 = zero8();
        gemm2<128>(Ax, 132, Wk, 768, 256 + j0, ln, hi, p0, p1);
        gemm2<256>(Ah, 260, Wr, 768, 256 + j0, ln, hi, q0, q1);
        {
            const float b0 = bias[256 + n], b1 = bias[768 + 256 + n];
#pragma unroll
            for (int v = 0; v < 8; ++v) {
                r0[v] = fast_sigmoid(p0[v] + b0 + q0[v] + b1);
                r1[v] = fast_sigmoid(p1[v] + b0 + q1[v] + b1);
            }
        }
        // ---- candidate + blend
        p0 = zero8(); p1 = zero8(); q0 = zero8(); q1 = zero8();
        gemm2<128>(Ax, 132, Wk, 768, 512 + j0, ln, hi, p0, p1);
        gemm2<256>(Ah, 260, Wr, 768, 512 + j0, ln, hi, q0, q1);
        {
            const float b0 = bias[512 + n], b1 = bias[768 + 512 + n];
#pragma unroll
            for (int v = 0; v < 8; ++v) {
                const int m = v + 8 * hi;
                float hh0 = fast_tanh(p0[v] + b0 + r0[v] * (q0[v] + b1));
                float hh1 = fast_tanh(p1[v] + b0 + r1[v] * (q1[v] + b1));
                float ho0 = Ah[m * 260 + n];
                float ho1 = Ah[(m + 16) * 260 + n];
                hro[(size_t)(row0 + m) * U_ + n]      = z0[v] * ho0 + (1.0f - z0[v]) * hh0;
                hro[(size_t)(row0 + 16 + m) * U_ + n] = z1[v] * ho1 + (1.0f - z1[v]) * hh1;
            }
        }
    }
}

// ---------------------------------------------------------------------------
// Kernel 4: column-wise partial sums of hro (first N_ rows only) into pooled.
// ---------------------------------------------------------------------------
__global__ __launch_bounds__(256)
void k_pool(const float* __restrict__ hro, float* __restrict__ pooled, int rowsPerBlk)
{
    const int c = threadIdx.x;
    int rbeg = blockIdx.x * rowsPerBlk;
    int rend = rbeg + rowsPerBlk;
    if (rend > N_) rend = N_;
    float sum = 0.0f;
    for (int r = rbeg; r < rend; ++r)
        sum += hro[(size_t)r * U_ + c];
    atomicAdd(&pooled[c], sum);
}

// ---------------------------------------------------------------------------
// Kernel 5: out = dot(pooled, W_ro) + b_ro
// ---------------------------------------------------------------------------
__global__ __launch_bounds__(256)
void k_final(const float* __restrict__ pooled, const float* __restrict__ Wro,
             const float* __restrict__ bro, float* __restrict__ out)
{
    __shared__ float red[256];
    red[threadIdx.x] = pooled[threadIdx.x] * Wro[threadIdx.x];
    __syncthreads();
    for (int s = 128; s > 0; s >>= 1) {
        if (threadIdx.x < s) red[threadIdx.x] += red[threadIdx.x + s];
        __syncthreads();
    }
    if (threadIdx.x == 0) out[0] = red[0] + bro[0];
}

// ---------------------------------------------------------------------------
extern "C" void kernel_launch(void* const* d_in, const int* in_sizes, int n_in,
                              void* d_out, int out_size, void* d_ws, size_t ws_size,
                              hipStream_t stream)
{
    (void)in_sizes; (void)n_in; (void)out_size; (void)ws_size;

    const float* link_state = (const float*)d_in[0];
    const int*   fidx       = (const int*)  d_in[1];
    const int*   sidx       = (const int*)  d_in[2];
    // d_in[3] = num_edges_critic (scalar), compile-time constant here
    const float* Wmsg       = (const float*)d_in[4];
    const float* bmsg       = (const float*)d_in[5];
    const float* updK       = (const float*)d_in[6];
    const float* updR       = (const float*)d_in[7];
    const float* updB       = (const float*)d_in[8];
    const float* gruK       = (const float*)d_in[9];
    const float* gruR       = (const float*)d_in[10];
    const float* gruB       = (const float*)d_in[11];
    const float* Wro        = (const float*)d_in[12];
    const float* bro        = (const float*)d_in[13];
    float*       out        = (float*)d_out;

    // workspace (floats): h[NP*128] | agg[NP*128] | hro[NP*256] | pooled[256]
    float* h      = (float*)d_ws;
    float* agg    = h   + (size_t)NP_ * D_;
    float* hro    = agg + (size_t)NP_ * D_;
    float* pooled = hro + (size_t)NP_ * U_;

    (void)hipMemcpyAsync(h, link_state, (size_t)N_ * D_ * sizeof(float),
                         hipMemcpyDeviceToDevice, stream);
    (void)hipMemsetAsync(h + (size_t)N_ * D_, 0,
                         (size_t)(NP_ - N_) * D_ * sizeof(float), stream);
    (void)hipMemsetAsync(hro, 0, (size_t)NP_ * U_ * sizeof(float), stream);
    (void)hipMemsetAsync(pooled, 0, U_ * sizeof(float), stream);

    const int edgeTiles = E_ / 16;     // 50000
    const int rowBlks   = NP_ / 32;    // 1563
    const size_t msgSmem = MSG_SMEM_FLOATS * sizeof(float);

    for (int step = 0; step < T_; ++step) {
        (void)hipMemsetAsync(agg, 0, (size_t)NP_ * D_ * sizeof(float), stream);
        k_msg    <<<1024,    256, msgSmem, stream>>>(h, fidx, sidx, Wmsg, bmsg, agg, edgeTiles);
        k_update <<<rowBlks, 256, 0, stream>>>(h, agg, updK, updR, updB);
        k_readout<<<rowBlks, 256, 0, stream>>>(hro, h, gruK, gruR, gruB);
    }

    const int rowsPerBlk = (N_ + 511) / 512;  // 98
    k_pool <<<512, 256, 0, stream>>>(hro, pooled, rowsPerBlk);
    k_final<<<1,   256, 0, stream>>>(pooled, Wro, bro, out);
}